// VolumeRendererModule_54838142435394
// MI455X (gfx1250) — compile-verified
//
#include <hip/hip_runtime.h>
#include <math.h>

typedef __attribute__((ext_vector_type(2))) float v2f;
typedef __attribute__((ext_vector_type(8))) float v8f;

#define WPB 8                       // waves (rays) per block
#define LDS_PER_WAVE (16*28 + 48)   // 16 samples x 28 params + 48 rgb results

__device__ __forceinline__ void wave_sync_lds() {
  __builtin_amdgcn_wave_barrier();
  asm volatile("s_wait_dscnt 0" ::: "memory");
  __builtin_amdgcn_wave_barrier();
}

__global__ __launch_bounds__(WPB * 32)
void volrend_wmma(const float* __restrict__ W, const float* __restrict__ rays,
                  float* __restrict__ out, int N)
{
  __shared__ alignas(16) float lds[WPB * LDS_PER_WAVE];
  const int  lane = threadIdx.x & 31;
  const bool hi   = lane >= 16;
  const int  wave = threadIdx.x >> 5;
  const int  ray  = blockIdx.x * WPB + wave;
  if (ray >= N) return;

  float* P = lds + wave * LDS_PER_WAVE;  // params: sample-major [16][28]
  float* R = P + 16 * 28;                // rgb results: [48] = (s,c) pairs

  // ---- per-ray setup (wave-uniform) ----
  const float* rp = rays + (size_t)ray * 6;
  const float ox = rp[0], oy = rp[1], oz = rp[2];
  const float xr = rp[3], yr = rp[4], zr = rp[5];   // d_raw
  const float nrm = sqrtf(xr*xr + yr*yr + zr*zr);
  const float dx = xr/nrm, dy = yr/nrm, dz = zr/nrm;

  const float ivx = 1.0f/dx, ivy = 1.0f/dy, ivz = 1.0f/dz;
  const float t0x = (-1.5f-ox)*ivx, t1x = (1.5f-ox)*ivx;
  const float t0y = (-1.5f-oy)*ivy, t1y = (1.5f-oy)*ivy;
  const float t0z = (-1.5f-oz)*ivz, t1z = (1.5f-oz)*ivz;
  const float tmin = fmaxf(fmaxf(fminf(t0x,t1x), fminf(t0y,t1y)), fminf(t0z,t1z));
  const float tmax = fminf(fminf(fmaxf(t0x,t1x), fmaxf(t0y,t1y)), fmaxf(t0z,t1z));
  const float nearT = fmaxf(tmin, 0.0f);
  const bool  isect = tmax >= nearT;
  const float span  = tmax - nearT;
  const int   ns    = isect ? (int)fminf(span * 32.0f, 256.0f) : 0;
  const float nsf   = (float)(ns > 1 ? ns : 1);
  const float dist  = span / nsf;

  // SH basis from the UNNORMALIZED direction (matches reference)
  const float b0 = 1.0f, b1 = xr, b2 = yr, b3 = zr;
  const float b4 = xr*yr, b5 = xr*zr, b6 = yr*zr, b7 = xr*xr, b8 = yr*yr;

  // B (4x16) replicated across all 16 columns; K split into chunks of 4.
  // K-row striping mirrors A/C/D: lo lanes hold rows {4kc,4kc+1}, hi lanes {4kc+2,4kc+3}.
  v2f B0, B1, B2;
  B0[0] = hi ? b2 : b0;   B0[1] = hi ? b3 : b1;
  B1[0] = hi ? b6 : b4;   B1[1] = hi ? b7 : b5;
  B2[0] = hi ? 0.0f : b8; B2[1] = 0.0f;

  float cx = 1.f, cy = 1.f, cz = 1.f, cw = 1.f;   // running color (1,1,1,1)
  const int nChunks = (ns + 15) >> 4;             // wave-uniform
  const int sl = lane & 15;
  const int kb = hi ? 2 : 0;

  for (int ch = 0; ch < nChunks; ++ch) {
    // ---- phase 1: gather voxel params into LDS (2 lanes per sample) ----
    const int   s  = ch * 16 + sl;
    const float t  = nearT + span * (((float)s + 0.5f) / nsf);
    const float px = (ox + dx*t) / 1.5f * 0.5f + 0.5f;
    const float py = (oy + dy*t) / 1.5f * 0.5f + 0.5f;
    const float pz = (oz + dz*t) / 1.5f * 0.5f + 0.5f;
    int ix = (int)floorf(px * 128.0f); ix = ix < 0 ? 0 : (ix > 127 ? 127 : ix);
    int iy = (int)floorf(py * 128.0f); iy = iy < 0 ? 0 : (iy > 127 ? 127 : iy);
    int iz = (int)floorf(pz * 128.0f); iz = iz < 0 ? 0 : (iz > 127 ? 127 : iz);
    const float4* vox = (const float4*)(W + (size_t)((ix*128 + iy)*128 + iz) * 28);
    const bool  v  = s < ns;
    const float4 zz = make_float4(0.f, 0.f, 0.f, 0.f);
    if (!hi) {
      float4 q0 = zz, q1 = zz, q2 = zz, q3 = zz;
      if (v) { q0 = vox[0]; q1 = vox[1]; q2 = vox[2]; q3 = vox[3]; }
      float4* dst = (float4*)(P + sl * 28);
      dst[0] = q0; dst[1] = q1; dst[2] = q2; dst[3] = q3;
    } else {
      float4 q4 = zz, q5 = zz, q6 = zz;
      if (v) { q4 = vox[4]; q5 = vox[5]; q6 = vox[6]; }
      float4* dst = (float4*)(P + sl * 28 + 16);
      dst[0] = q4; dst[1] = q5; dst[2] = q6;
    }
    wave_sync_lds();

    // ---- phase 2: SH matvec via WMMA f32 16x16x4 ----
    // Tile rows r = (sample,channel) pairs; pair = 16*tile + r; rgb[pair] = sum_k sh*b.
    v8f D0, D1, D2;
#pragma unroll
    for (int tt = 0; tt < 3; ++tt) {
      const int pair = tt * 16 + sl;
      const int ss = pair / 3;
      const int cc = pair - ss * 3;
      const float* shp = P + ss * 28 + cc * 9;
      v2f A0, A1, A2;
      A0[0] = shp[kb + 0]; A0[1] = shp[kb + 1];
      A1[0] = shp[kb + 4]; A1[1] = shp[kb + 5];
      A2[0] = hi ? 0.0f : shp[8]; A2[1] = 0.0f;
      v8f acc = {0.f,0.f,0.f,0.f,0.f,0.f,0.f,0.f};
      acc = __builtin_amdgcn_wmma_f32_16x16x4_f32(false, A0, false, B0, (short)0, acc, false, false);
      acc = __builtin_amdgcn_wmma_f32_16x16x4_f32(false, A1, false, B1, (short)0, acc, false, false);
      acc = __builtin_amdgcn_wmma_f32_16x16x4_f32(false, A2, false, B2, (short)0, acc, false, false);
      if (tt == 0) D0 = acc; else if (tt == 1) D1 = acc; else D2 = acc;
    }
    // D rows: vgpr i = row i (lanes 0-15) / row 8+i (lanes 16-31); columns identical.
    if (lane == 0 || lane == 16) {
      const int ro = hi ? 8 : 0;
#pragma unroll
      for (int i = 0; i < 8; ++i) {
        R[ 0 + ro + i] = D0[i];
        R[16 + ro + i] = D1[i];
        R[32 + ro + i] = D2[i];
      }
    }
    wave_sync_lds();

    // ---- phase 3: alpha + affine-scan composite over the 16 samples ----
    const bool  valid = s < ns;
    const float sigma = P[sl * 28 + 27];
    const float rr = R[3*sl + 0], rg = R[3*sl + 1], rb = R[3*sl + 2];
    const float a  = valid ? (1.0f - expf(sigma * dist)) : 1.0f;
    const float w1 = 1.0f - a;
    float m  = a;
    float qx = valid ? rr * w1    : 0.0f;
    float qy = valid ? rg * w1    : 0.0f;
    float qz = valid ? rb * w1    : 0.0f;
    float qw = valid ? sigma * w1 : 0.0f;
#pragma unroll
    for (int off = 1; off < 16; off <<= 1) {
      const float pm  = __shfl_up(m,  off);
      const float pqx = __shfl_up(qx, off);
      const float pqy = __shfl_up(qy, off);
      const float pqz = __shfl_up(qz, off);
      const float pqw = __shfl_up(qw, off);
      if (sl >= off) {
        qx = pqx * m + qx; qy = pqy * m + qy;
        qz = pqz * m + qz; qw = pqw * m + qw;
        m  = pm * m;
      }
    }
    const float M  = __shfl(m, 15);
    const float FX = __shfl(qx, 15);
    const float FY = __shfl(qy, 15);
    const float FZ = __shfl(qz, 15);
    const float FW = __shfl(qw, 15);
    cx = cx * M + FX; cy = cy * M + FY; cz = cz * M + FZ; cw = cw * M + FW;

    wave_sync_lds();
  }

  if (lane == 0) {
    *(float4*)(out + (size_t)ray * 4) = make_float4(cx, cy, cz, cw);
  }
}

extern "C" void kernel_launch(void* const* d_in, const int* in_sizes, int n_in,
                              void* d_out, int out_size, void* d_ws, size_t ws_size,
                              hipStream_t stream) {
  const float* W    = (const float*)d_in[0];   // (128,128,128,28) f32
  const float* rays = (const float*)d_in[1];   // (N,6) f32
  float* out = (float*)d_out;                  // (N,4) f32
  const int N = in_sizes[1] / 6;
  const int blocks = (N + WPB - 1) / WPB;
  volrend_wmma<<<blocks, WPB * 32, 0, stream>>>(W, rays, out, N);
}